// EpisodicMemory_66185446031745
// MI455X (gfx1250) — compile-verified
//
#include <hip/hip_runtime.h>
#include <cstdint>
#include <cstddef>

// ---------------------------------------------------------------------------
// EpisodicMemory fused kernel for gfx1250 (MI455X), fp32 via V_WMMA_F32_16X16X4_F32
// ---------------------------------------------------------------------------

typedef __attribute__((ext_vector_type(2))) float v2f;
typedef __attribute__((ext_vector_type(8))) float v8f;

#define D_DIM 512
#define LDA   516    // LDS row stride (floats): 516 mod 64 banks -> 4-bank rotation per row
#define CHS   260    // sim chunk LDS row stride
#define CHUNK 256    // keys per sim chunk (4 chunks x 256 = 1024)

static __device__ __forceinline__ v8f vzero8() {
  v8f z = {0.f, 0.f, 0.f, 0.f, 0.f, 0.f, 0.f, 0.f};
  return z;
}

// D = A(16x4) * B(4x16) + C, fp32. 8-arg form: (neg_a, A, neg_b, B, c_mod, C, reuse_a, reuse_b)
static __device__ __forceinline__ v8f wmma4(v2f a, v2f b, v8f c) {
  return __builtin_amdgcn_wmma_f32_16x16x4_f32(false, a, false, b, (short)0, c, false, false);
}

// K=512 dot chain. Arow/Brow already offset to (row, 2*half); both row-major.
static __device__ __forceinline__ v8f gemm_k512(const float* __restrict__ Arow,
                                                const float* __restrict__ Brow,
                                                v8f acc) {
  __builtin_prefetch(Brow, 0, 3);
  #pragma unroll 4
  for (int k = 0; k < D_DIM; k += 4) {
    v2f a = *(const v2f*)(Arow + k);
    v2f b = *(const v2f*)(Brow + k);
    acc = wmma4(a, b, acc);
  }
  return acc;
}

// Scatter a 16x16 f32 D-tile into LDS: VGPR j -> row (j + 8*half), col n0 + (lane&15)
static __device__ __forceinline__ void store_tile(float* dst, int stride, int n0,
                                                  int lane, v8f acc) {
  const int half = lane >> 4;
  const int col  = n0 + (lane & 15);
  #pragma unroll
  for (int j = 0; j < 8; ++j) dst[(j + 8 * half) * stride + col] = acc[j];
}

// ---------------- per-lane running top-4 ----------------
struct Top4 { float a, b, c, d; int ia, ib, ic, id; };

static __device__ __forceinline__ void t4_init(Top4& t) {
  t.a = t.b = t.c = t.d = -1e30f;
  t.ia = t.ib = t.ic = t.id = 0x7fffffff;
}

static __device__ __forceinline__ void t4_insert(Top4& t, float s, int e) {
  if (s > t.d) {
    if (s > t.b) {
      if (s > t.a) { t.d=t.c; t.id=t.ic; t.c=t.b; t.ic=t.ib; t.b=t.a; t.ib=t.ia; t.a=s; t.ia=e; }
      else         { t.d=t.c; t.id=t.ic; t.c=t.b; t.ic=t.ib; t.b=s; t.ib=e; }
    } else {
      if (s > t.c) { t.d=t.c; t.id=t.ic; t.c=s; t.ic=e; }
      else         { t.d=s; t.id=e; }
    }
  }
}

static __device__ __forceinline__ void t4_pop(Top4& t) {
  t.a=t.b; t.ia=t.ib; t.b=t.c; t.ib=t.ic; t.c=t.d; t.ic=t.id;
  t.d = -1e30f; t.id = 0x7fffffff;
}

// cross-lane merge of per-lane sorted top-4 lists + softmax(5x) -> sWgt/sIdx[row]
static __device__ __forceinline__ void finalize_row(Top4 t, int m, int lane,
                                                    float* sWgt, int* sIdx) {
  float tv[4]; int ti[4];
  #pragma unroll
  for (int kk = 0; kk < 4; ++kk) {
    float bv = t.a; int bi = t.ia;
    #pragma unroll
    for (int off = 16; off > 0; off >>= 1) {
      float ov = __shfl_xor(bv, off, 32);
      int   oi = __shfl_xor(bi, off, 32);
      if (ov > bv || (ov == bv && oi < bi)) { bv = ov; bi = oi; }
    }
    tv[kk] = bv; ti[kk] = bi;
    if (t.ia == bi) t4_pop(t);   // unique owner pops its head
  }
  const float e1 = __expf(5.f * (tv[1] - tv[0]));
  const float e2 = __expf(5.f * (tv[2] - tv[0]));
  const float e3 = __expf(5.f * (tv[3] - tv[0]));
  const float inv = 1.f / (1.f + e1 + e2 + e3);
  if (lane == 0) {
    sWgt[m*4+0] = inv;      sWgt[m*4+1] = e1 * inv;
    sWgt[m*4+2] = e2 * inv; sWgt[m*4+3] = e3 * inv;
    sIdx[m*4+0] = ti[0]; sIdx[m*4+1] = ti[1];
    sIdx[m*4+2] = ti[2]; sIdx[m*4+3] = ti[3];
  }
}

// ---------------------------------------------------------------------------
// Kernel 1: keys_norm[e] = episode_keys[e] / max(||.||, eps)  (E rows of 512)
// ---------------------------------------------------------------------------
__global__ __launch_bounds__(256)
void ep_norm_keys(const float* __restrict__ K, float* __restrict__ KN) {
  const int w = threadIdx.x >> 5, lane = threadIdx.x & 31;
  const int row = blockIdx.x * 8 + w;
  const float* src = K  + (size_t)row * D_DIM;
  float*       dst = KN + (size_t)row * D_DIM;
  float buf[16];
  float s = 0.f;
  #pragma unroll
  for (int i = 0; i < 16; ++i) { float v = src[lane + 32*i]; buf[i] = v; s += v*v; }
  #pragma unroll
  for (int off = 16; off > 0; off >>= 1) s += __shfl_xor(s, off, 32);
  const float inv = 1.f / fmaxf(sqrtf(s), 1e-12f);
  #pragma unroll
  for (int i = 0; i < 16; ++i) dst[lane + 32*i] = buf[i] * inv;
}

// ---------------------------------------------------------------------------
// Kernel 2: fully fused per 16-row tile of B:
//   qproj -> normalize -> sim(+topk, chunked) -> softmax -> retrieve
//   -> vproj -> gate -> out
// ---------------------------------------------------------------------------
__global__ __launch_bounds__(256)
void ep_fused(const float* __restrict__ query, const float* __restrict__ keysn,
              const float* __restrict__ evals, const float* __restrict__ Wq,
              const float* __restrict__ Wv,    const float* __restrict__ Wg,
              const float* __restrict__ bg,    float* __restrict__ out) {
  __shared__ float sA[16 * LDA];     // q_norm, then retrieved, then projected
  __shared__ float sC[16 * CHS];     // sim chunk
  __shared__ float sWgt[16 * 4];
  __shared__ int   sIdx[16 * 4];

  const int tid  = threadIdx.x;
  const int w    = tid >> 5;
  const int lane = tid & 31;
  const int half = lane >> 4;
  const int r    = lane & 15;
  const int row0 = blockIdx.x * 16;

  // ---- Phase 1: q = query @ Wq^T  (wave w owns cols [w*64, w*64+64)) ----
  const float* Aq = query + (size_t)(row0 + r) * D_DIM + 2 * half;
  #pragma unroll
  for (int t = 0; t < 4; ++t) {
    const int n0 = w * 64 + t * 16;
    v8f acc = gemm_k512(Aq, Wq + (size_t)(n0 + r) * D_DIM + 2 * half, vzero8());
    store_tile(sA, LDA, n0, lane, acc);
  }
  __syncthreads();

  // ---- Phase 2: L2-normalize rows of q in LDS (wave w -> rows 2w, 2w+1) ----
  #pragma unroll
  for (int mi = 0; mi < 2; ++mi) {
    const int m = 2 * w + mi;
    float s = 0.f;
    #pragma unroll
    for (int i = 0; i < 16; ++i) { float v = sA[m * LDA + lane + 32*i]; s += v*v; }
    #pragma unroll
    for (int off = 16; off > 0; off >>= 1) s += __shfl_xor(s, off, 32);
    const float inv = 1.f / fmaxf(sqrtf(s), 1e-12f);
    #pragma unroll
    for (int i = 0; i < 16; ++i) sA[m * LDA + lane + 32*i] *= inv;
  }
  __syncthreads();

  // ---- Phase 3: sim = q_norm @ keys_norm^T, chunked; running per-lane top4 ----
  Top4 ta, tb; t4_init(ta); t4_init(tb);
  const float* Asim = sA + r * LDA + 2 * half;
  for (int c = 0; c < 4; ++c) {
    const int e0 = c * CHUNK;
    #pragma unroll
    for (int t = 0; t < 2; ++t) {
      const int n0 = w * 32 + t * 16;  // column within chunk
      v8f acc = gemm_k512(Asim, keysn + (size_t)(e0 + n0 + r) * D_DIM + 2 * half, vzero8());
      store_tile(sC, CHS, n0, lane, acc);
    }
    __syncthreads();
    #pragma unroll
    for (int i = 0; i < CHUNK / 32; ++i) {
      const int el = lane + 32 * i;
      t4_insert(ta, sC[(2*w + 0) * CHS + el], e0 + el);
      t4_insert(tb, sC[(2*w + 1) * CHS + el], e0 + el);
    }
    __syncthreads();
  }
  finalize_row(ta, 2*w + 0, lane, sWgt, sIdx);
  finalize_row(tb, 2*w + 1, lane, sWgt, sIdx);
  __syncthreads();

  // ---- Phase 4: retrieved[m] = sum_j wgt[m][j] * evals[idx[m][j]] -> sA ----
  for (int i = tid; i < 16 * (D_DIM / 4); i += 256) {
    const int m  = i >> 7;
    const int c4 = (i & 127) << 2;
    const float w0 = sWgt[m*4+0], w1 = sWgt[m*4+1], w2 = sWgt[m*4+2], w3 = sWgt[m*4+3];
    const float4 x0 = *(const float4*)(evals + (size_t)sIdx[m*4+0] * D_DIM + c4);
    const float4 x1 = *(const float4*)(evals + (size_t)sIdx[m*4+1] * D_DIM + c4);
    const float4 x2 = *(const float4*)(evals + (size_t)sIdx[m*4+2] * D_DIM + c4);
    const float4 x3 = *(const float4*)(evals + (size_t)sIdx[m*4+3] * D_DIM + c4);
    sA[m*LDA + c4 + 0] = w0*x0.x + w1*x1.x + w2*x2.x + w3*x3.x;
    sA[m*LDA + c4 + 1] = w0*x0.y + w1*x1.y + w2*x2.y + w3*x3.y;
    sA[m*LDA + c4 + 2] = w0*x0.z + w1*x1.z + w2*x2.z + w3*x3.z;
    sA[m*LDA + c4 + 3] = w0*x0.w + w1*x1.w + w2*x2.w + w3*x3.w;
  }
  __syncthreads();

  // ---- Phase 5: projected = retrieved @ Wv^T (regs, then write back to sA) ----
  v8f pacc[4];
  const float* Ar = sA + r * LDA + 2 * half;
  #pragma unroll
  for (int t = 0; t < 4; ++t) {
    const int n0 = w * 64 + t * 16;
    pacc[t] = gemm_k512(Ar, Wv + (size_t)(n0 + r) * D_DIM + 2 * half, vzero8());
  }
  __syncthreads();   // everyone done reading retrieved
  #pragma unroll
  for (int t = 0; t < 4; ++t) store_tile(sA, LDA, w * 64 + t * 16, lane, pacc[t]);
  __syncthreads();   // projected visible block-wide

  // ---- Phase 6: gate = sigmoid(query@Wg1^T + projected@Wg2^T + bg); out = gate*projected
  #pragma unroll
  for (int t = 0; t < 4; ++t) {
    const int n0 = w * 64 + t * 16;
    v8f acc = gemm_k512(Aq, Wg + (size_t)(n0 + r) * (2 * D_DIM) + 2 * half, vzero8());
    acc = gemm_k512(sA + r * LDA + 2 * half,
                    Wg + (size_t)(n0 + r) * (2 * D_DIM) + D_DIM + 2 * half, acc);
    const int col = n0 + r;
    const float bgv = bg[col];
    #pragma unroll
    for (int j = 0; j < 8; ++j) {
      const int m = j + 8 * half;
      const float g = 1.f / (1.f + __expf(-(acc[j] + bgv)));
      out[(size_t)(row0 + m) * D_DIM + col] = g * sA[m * LDA + col];
    }
  }
}

// ---------------------------------------------------------------------------
extern "C" void kernel_launch(void* const* d_in, const int* in_sizes, int n_in,
                              void* d_out, int out_size, void* d_ws, size_t ws_size,
                              hipStream_t stream) {
  (void)n_in; (void)out_size; (void)ws_size;
  const float* query = (const float*)d_in[0];
  const float* ekeys = (const float*)d_in[1];
  const float* evals = (const float*)d_in[2];
  const float* Wq    = (const float*)d_in[3];
  const float* Wv    = (const float*)d_in[4];
  const float* Wg    = (const float*)d_in[5];
  const float* bg    = (const float*)d_in[6];
  float* out   = (float*)d_out;
  float* keysn = (float*)d_ws;                 // E*D f32 = 2 MB scratch

  const int B = in_sizes[0] / D_DIM;           // 16384
  const int E = in_sizes[1] / D_DIM;           // 1024

  hipLaunchKernelGGL(ep_norm_keys, dim3(E / 8), dim3(256), 0, stream, ekeys, keysn);
  hipLaunchKernelGGL(ep_fused, dim3(B / 16), dim3(256), 0, stream,
                     query, keysn, evals, Wq, Wv, Wg, bg, out);
}